// CustomDeepGPRGNN_9955734192491
// MI455X (gfx1250) — compile-verified
//
#include <hip/hip_runtime.h>
#include <hip/hip_bf16.h>
#include <math.h>

typedef float v2f __attribute__((ext_vector_type(2)));
typedef float v8f __attribute__((ext_vector_type(8)));

#define THREADS 256
#define HDIM 256
#define INDIM 512
#define OUTDIM 64
#define KSTEPS 10
#define BN_EPS 1e-5f
#define INV_TEMP (1.0f / 1.5f)

static inline int cdiv_i(long long a, long long b) { return (int)((a + b - 1) / b); }

// ---------------- generic fill ----------------
__global__ void fill_kernel(float* __restrict__ p, float v, int n) {
  int i = blockIdx.x * blockDim.x + threadIdx.x;
  if (i < n) p[i] = v;
}

// ---------------- pack weight matrix into WMMA B-fragment order ----------------
// Bp[ct][q][lane] = float4{ B[4q+2h][ct*32+l], B[4q+2h+1][ct*32+l],
//                           B[4q+2h][ct*32+16+l], B[4q+2h+1][ct*32+16+l] }
// with l = lane&15, h = lane>>4. One b128 load then supplies both 4x16 B fragments.
__global__ void pack_b(const float* __restrict__ B, float4* __restrict__ Bp, int K, int N) {
  int t = blockIdx.x * blockDim.x + threadIdx.x;
  int total = (N / 32) * (K / 4) * 32;
  if (t >= total) return;
  int lane = t & 31;
  int q    = (t >> 5) % (K / 4);
  int ct   = (t >> 5) / (K / 4);
  int l = lane & 15, half = lane >> 4;
  int k = 4 * q + 2 * half;
  int c = ct * 32 + l;
  float4 v;
  v.x = B[(size_t)k * N + c];
  v.y = B[(size_t)(k + 1) * N + c];
  v.z = B[(size_t)k * N + c + 16];
  v.w = B[(size_t)(k + 1) * N + c + 16];
  Bp[t] = v;
}

// ---------------- fp32 WMMA GEMM: C[M,NC] = A[M,K] @ B[K,NC] (+bias) ----------------
// One wave computes a 32x32 tile as 2x2 fragments of V_WMMA_F32_16X16X4_F32.
// K, NC compile-time: strides fold into immediate offsets; waves in a block share
// the column tile (packed-B lines hit L0) and stream distinct A row-tiles.
template <int K, int NC>
__global__ void gemm_wmma(const float* __restrict__ A, const float* __restrict__ Bp,
                          const float* __restrict__ bias, float* __restrict__ C,
                          int totalTiles, int rowTiles) {
  int wid = blockIdx.x * (blockDim.x >> 5) + (threadIdx.x >> 5);
  if (wid >= totalTiles) return;          // wave-uniform: EXEC stays all-ones for WMMA
  int rt = wid % rowTiles;
  int ct = wid / rowTiles;
  int lane = threadIdx.x & 31;
  int l    = lane & 15;
  int half = lane >> 4;

  const float*  Arow0 = A + (size_t)(rt * 32 + l) * K + half * 2;
  const float*  Arow1 = Arow0 + (size_t)16 * K;
  const float4* bp    = (const float4*)Bp + ((size_t)ct * (K / 4)) * 32 + lane;

  v8f acc00 = {}, acc01 = {}, acc10 = {}, acc11 = {};
#pragma unroll 4
  for (int q = 0; q < K / 4; ++q) {
    v2f a0 = *(const v2f*)(Arow0 + 4 * q);
    v2f a1 = *(const v2f*)(Arow1 + 4 * q);
    float4 b = bp[q * 32];
    v2f b0; b0.x = b.x; b0.y = b.y;
    v2f b1; b1.x = b.z; b1.y = b.w;
    acc00 = __builtin_amdgcn_wmma_f32_16x16x4_f32(false, a0, false, b0, (short)0, acc00, false, false);
    acc01 = __builtin_amdgcn_wmma_f32_16x16x4_f32(false, a0, false, b1, (short)0, acc01, false, false);
    acc10 = __builtin_amdgcn_wmma_f32_16x16x4_f32(false, a1, false, b0, (short)0, acc10, false, false);
    acc11 = __builtin_amdgcn_wmma_f32_16x16x4_f32(false, a1, false, b1, (short)0, acc11, false, false);
  }

  float bias0 = 0.f, bias1 = 0.f;
  if (bias) { bias0 = bias[ct * 32 + l]; bias1 = bias[ct * 32 + 16 + l]; }

  int row0 = rt * 32 + half * 8;
  int col0 = ct * 32 + l;
#pragma unroll
  for (int r = 0; r < 8; ++r) {
    C[(size_t)(row0 + r) * NC + col0]           = acc00[r] + bias0;
    C[(size_t)(row0 + r) * NC + col0 + 16]      = acc01[r] + bias1;
    C[(size_t)(row0 + 16 + r) * NC + col0]      = acc10[r] + bias0;
    C[(size_t)(row0 + 16 + r) * NC + col0 + 16] = acc11[r] + bias1;
  }
}

// ---------------- BatchNorm column statistics ----------------
__global__ void bn_stats_partial(const float* __restrict__ h, int Nrows,
                                 float* __restrict__ sums /* [2*HDIM] */) {
  int c  = threadIdx.x;                 // blockDim.x == HDIM
  int r0 = blockIdx.x * 512;
  int r1 = r0 + 512; if (r1 > Nrows) r1 = Nrows;
  float s = 0.f, s2 = 0.f;
  for (int r = r0; r < r1; ++r) {
    float v = h[(size_t)r * HDIM + c];
    s += v; s2 += v * v;
  }
  atomicAdd(&sums[c], s);
  atomicAdd(&sums[HDIM + c], s2);
}

__global__ void bn_finalize(const float* __restrict__ sums, int Nrows,
                            float* __restrict__ meanv, float* __restrict__ rstd) {
  int c = threadIdx.x;
  if (c < HDIM) {
    float inv_n = 1.0f / (float)Nrows;
    float m = sums[c] * inv_n;
    float v = sums[HDIM + c] * inv_n - m * m;
    meanv[c] = m;
    rstd[c]  = rsqrtf(v + BN_EPS);
  }
}

// ---------------- BN + (residual) + exact GELU ----------------
__global__ void bn_gelu(const float* __restrict__ h, const float* __restrict__ res,
                        const float* __restrict__ meanv, const float* __restrict__ rstd,
                        const float* __restrict__ g, const float* __restrict__ be,
                        float* __restrict__ y, int total) {
  int i = blockIdx.x * blockDim.x + threadIdx.x;
  if (i < total) {
    int c = i & (HDIM - 1);
    float t = g[c] * (h[i] - meanv[c]) * rstd[c] + be[c];
    if (res) t += res[i];
    y[i] = 0.5f * t * (1.0f + erff(t * 0.70710678118654752f));
  }
}

// ---------------- graph normalization ----------------
__global__ void deg_accum(const int* __restrict__ row, int E, float* __restrict__ deg) {
  int e = blockIdx.x * blockDim.x + threadIdx.x;
  if (e < E) atomicAdd(&deg[row[e]], 1.0f);
}

__global__ void dinv_kernel(const float* __restrict__ deg, float* __restrict__ dinv,
                            float* __restrict__ selfn, int n) {
  int i = blockIdx.x * blockDim.x + threadIdx.x;
  if (i < n) {
    float d  = deg[i];
    float di = d > 0.f ? rsqrtf(d) : 0.f;
    dinv[i]  = di;
    selfn[i] = di * di;
  }
}

__global__ void norm_kernel(const int* __restrict__ row, const int* __restrict__ col,
                            const float* __restrict__ dinv, float* __restrict__ nrm, int E) {
  int e = blockIdx.x * blockDim.x + threadIdx.x;
  if (e < E) nrm[e] = dinv[row[e]] * dinv[col[e]];
}

// ---------------- propagation ----------------
// Fused: acc (+)= (alpha[k]/TEMP)*cur  AND  nxt = selfnorm*cur (self-loop init for
// the scatter that follows). Saves a full extra pass over the feature matrix.
__global__ void prop_self_axpy(const float4* __restrict__ cur, const float* __restrict__ selfn,
                               float4* __restrict__ nxt, float4* __restrict__ acc,
                               const float* __restrict__ alpha, int k, int setMode, int n16) {
  int i = blockIdx.x * blockDim.x + threadIdx.x;
  if (i < n16) {
    float a = alpha[k] * INV_TEMP;
    float s = selfn[i >> 4];
    float4 v = cur[i];
    float4 o; o.x = s * v.x; o.y = s * v.y; o.z = s * v.z; o.w = s * v.w;
    nxt[i] = o;
    if (setMode) {
      float4 t; t.x = a * v.x; t.y = a * v.y; t.z = a * v.z; t.w = a * v.w;
      acc[i] = t;
    } else {
      float4 t = acc[i];
      t.x += a * v.x; t.y += a * v.y; t.z += a * v.z; t.w += a * v.w;
      acc[i] = t;
    }
  }
}

__global__ void prop_scatter(const int* __restrict__ row, const int* __restrict__ col,
                             const float* __restrict__ nrm, const float* __restrict__ cur,
                             float* __restrict__ nxt, int E) {
  int t = blockIdx.x * blockDim.x + threadIdx.x;
  int e = t >> 4;
  if (e >= E) return;
  int f4 = t & 15;
  int r = row[e], c = col[e];
  float w = nrm[e];
  float4 v = *((const float4*)(cur + (size_t)c * OUTDIM) + f4);
  float* dst = nxt + (size_t)r * OUTDIM + f4 * 4;
  atomicAdd(dst + 0, w * v.x);
  atomicAdd(dst + 1, w * v.y);
  atomicAdd(dst + 2, w * v.z);
  atomicAdd(dst + 3, w * v.w);
}

__global__ void axpy_acc(const float4* __restrict__ src, const float* __restrict__ alpha,
                         int k, float4* __restrict__ acc, int n4) {
  int i = blockIdx.x * blockDim.x + threadIdx.x;
  if (i < n4) {
    float a  = alpha[k] * INV_TEMP;
    float4 v = src[i];
    float4 o = acc[i];
    o.x += a * v.x; o.y += a * v.y; o.z += a * v.z; o.w += a * v.w;
    acc[i] = o;
  }
}

// ---------------- host orchestration ----------------
extern "C" void kernel_launch(void* const* d_in, const int* in_sizes, int n_in,
                              void* d_out, int out_size, void* d_ws, size_t ws_size,
                              hipStream_t stream) {
  const float* x    = (const float*)d_in[0];
  const int*   eidx = (const int*)d_in[1];
  const float* w1 = (const float*)d_in[2];
  const float* g1 = (const float*)d_in[4];  const float* be1 = (const float*)d_in[5];
  const float* w2 = (const float*)d_in[6];
  const float* g2 = (const float*)d_in[8];  const float* be2 = (const float*)d_in[9];
  const float* w3 = (const float*)d_in[10];
  const float* g3 = (const float*)d_in[12]; const float* be3 = (const float*)d_in[13];
  const float* w4 = (const float*)d_in[14]; const float* b4  = (const float*)d_in[15];
  const float* alpha = (const float*)d_in[16];
  float* out = (float*)d_out;

  const int NN = in_sizes[0] / INDIM;     // 100000
  const int E  = in_sizes[1] / 2;         // 3200000
  const int* row = eidx;
  const int* col = eidx + E;

  // workspace layout (256B aligned slabs)
  char*  wsb = (char*)d_ws;
  size_t off = 0;
  auto alloc = [&](size_t bytes) -> void* {
    void* p = wsb + off;
    off = (off + bytes + 255) & ~(size_t)255;
    return p;
  };
  float*  bufA  = (float*)alloc((size_t)NN * HDIM * sizeof(float));
  float*  bufB  = (float*)alloc((size_t)NN * HDIM * sizeof(float));
  float*  nrm   = (float*)alloc((size_t)E * sizeof(float));
  float*  deg   = (float*)alloc((size_t)NN * sizeof(float));
  float*  dinv  = (float*)alloc((size_t)NN * sizeof(float));
  float*  selfn = (float*)alloc((size_t)NN * sizeof(float));
  float*  sums  = (float*)alloc(2 * HDIM * sizeof(float));
  float*  meanv = (float*)alloc(HDIM * sizeof(float));
  float*  rstd  = (float*)alloc(HDIM * sizeof(float));
  float4* Bp    = (float4*)alloc((size_t)INDIM * HDIM * sizeof(float)); // max packed weight
  (void)ws_size; (void)n_in; (void)out_size;

  const int totalEl  = NN * HDIM;
  const int statBlks = cdiv_i(NN, 512);
  const int rowTiles = NN / 32;

  auto run_pack = [&](const float* B, int K, int Ncols) {
    int total = (Ncols / 32) * (K / 4) * 32;
    hipLaunchKernelGGL(pack_b, dim3(cdiv_i(total, THREADS)), dim3(THREADS), 0, stream,
                       B, Bp, K, Ncols);
  };
  auto run_bn = [&](const float* h, const float* res, const float* g, const float* be,
                    float* y) {
    hipLaunchKernelGGL(fill_kernel, dim3(cdiv_i(2 * HDIM, THREADS)), dim3(THREADS), 0, stream,
                       sums, 0.0f, 2 * HDIM);
    hipLaunchKernelGGL(bn_stats_partial, dim3(statBlks), dim3(HDIM), 0, stream, h, NN, sums);
    hipLaunchKernelGGL(bn_finalize, dim3(1), dim3(HDIM), 0, stream, sums, NN, meanv, rstd);
    hipLaunchKernelGGL(bn_gelu, dim3(cdiv_i(totalEl, THREADS)), dim3(THREADS), 0, stream,
                       h, res, meanv, rstd, g, be, y, totalEl);
  };

  // ---- MLP (BN cancels b1..b3 exactly; b4 applied in GEMM4 epilogue) ----
  {
    int tiles = rowTiles * (HDIM / 32);
    int blocks = cdiv_i((long long)tiles * 32, THREADS);
    run_pack(w1, INDIM, HDIM);
    hipLaunchKernelGGL((gemm_wmma<INDIM, HDIM>), dim3(blocks), dim3(THREADS), 0, stream,
                       x, (const float*)Bp, nullptr, bufB, tiles, rowTiles);
    run_bn(bufB, nullptr, g1, be1, bufA);            // x0 -> bufA

    run_pack(w2, HDIM, HDIM);
    hipLaunchKernelGGL((gemm_wmma<HDIM, HDIM>), dim3(blocks), dim3(THREADS), 0, stream,
                       bufA, (const float*)Bp, nullptr, bufB, tiles, rowTiles);
    run_bn(bufB, bufA, g2, be2, bufB);               // x1 -> bufB (in place)

    run_pack(w3, HDIM, HDIM);
    hipLaunchKernelGGL((gemm_wmma<HDIM, HDIM>), dim3(blocks), dim3(THREADS), 0, stream,
                       bufB, (const float*)Bp, nullptr, bufA, tiles, rowTiles);
    run_bn(bufA, bufB, g3, be3, bufA);               // x2 -> bufA (in place)

    int tiles4 = rowTiles * (OUTDIM / 32);
    int blocks4 = cdiv_i((long long)tiles4 * 32, THREADS);
    run_pack(w4, HDIM, OUTDIM);
    hipLaunchKernelGGL((gemm_wmma<HDIM, OUTDIM>), dim3(blocks4), dim3(THREADS), 0, stream,
                       bufA, (const float*)Bp, b4, bufB, tiles4, rowTiles);  // h -> bufB [NN,64]
  }

  // ---- graph normalization (deg counted on row, +1 self loop) ----
  hipLaunchKernelGGL(fill_kernel, dim3(cdiv_i(NN, THREADS)), dim3(THREADS), 0, stream,
                     deg, 1.0f, NN);
  hipLaunchKernelGGL(deg_accum, dim3(cdiv_i(E, THREADS)), dim3(THREADS), 0, stream,
                     row, E, deg);
  hipLaunchKernelGGL(dinv_kernel, dim3(cdiv_i(NN, THREADS)), dim3(THREADS), 0, stream,
                     deg, dinv, selfn, NN);
  hipLaunchKernelGGL(norm_kernel, dim3(cdiv_i(E, THREADS)), dim3(THREADS), 0, stream,
                     row, col, dinv, nrm, E);

  // ---- K-step propagation with fused alpha accumulation (scaled by 1/TEMP) ----
  const int n4  = NN * OUTDIM / 4;                 // float4 elements of acc
  const int n16 = NN * 16;                         // (node, feat4) pairs
  float* cur = bufB;
  float* nxt = bufA;
  for (int k = 0; k < KSTEPS; ++k) {
    hipLaunchKernelGGL(prop_self_axpy, dim3(cdiv_i(n16, THREADS)), dim3(THREADS), 0, stream,
                       (const float4*)cur, selfn, (float4*)nxt, (float4*)out,
                       alpha, k, (k == 0) ? 1 : 0, n16);
    hipLaunchKernelGGL(prop_scatter, dim3(cdiv_i((long long)E * 16, THREADS)), dim3(THREADS),
                       0, stream, row, col, nrm, cur, nxt, E);
    float* t = cur; cur = nxt; nxt = t;
  }
  hipLaunchKernelGGL(axpy_acc, dim3(cdiv_i(n4, THREADS)), dim3(THREADS), 0, stream,
                     (const float4*)cur, alpha, KSTEPS, (float4*)out, n4);
}